// HungarianMatchingLoss_21234318311796
// MI455X (gfx1250) — compile-verified
//
#include <hip/hip_runtime.h>
#include <stdint.h>

// Problem constants (fixed by the reference harness: B=32, Q=900).
#define BATCH 32
#define NQ 900
#define ROWS_PER_BLOCK 10
#define GEPS 1e-6f

typedef unsigned int u32x4 __attribute__((ext_vector_type(4)));
typedef int i32x8 __attribute__((ext_vector_type(8)));
typedef int i32x4 __attribute__((ext_vector_type(4)));

#if defined(__gfx1250__) && __has_builtin(__builtin_amdgcn_tensor_load_to_lds)
#define USE_TDM 1
#else
#define USE_TDM 0
#endif

// ---------------------------------------------------------------------------
// Phase 1: pairwise GIoU/IoU.
//   giou_ws[b,i,j]  -> workspace (kept L2-resident with default temporal hints;
//                      the 104MB tensor fits in MI455X's 192MB L2 and is the
//                      hot working set of the JV solver)
//   iou_out[b,i,j]  -> d_out+1, streamed with non-temporal stores (write-once)
// gt boxes of the batch (3600 floats = 14.4KB) are staged into LDS once per
// block via the Tensor Data Mover (CDNA5 tensor_load_to_lds).
// ---------------------------------------------------------------------------
__global__ __launch_bounds__(256) void giou_pairwise_kernel(
    const float* __restrict__ pred, const float* __restrict__ gt,
    const unsigned char* __restrict__ valid,
    float* __restrict__ giou_ws, float* __restrict__ iou_out) {
  __shared__ float gtile[NQ * 4];
  __shared__ unsigned char vtile[NQ];

  const int b = blockIdx.y;
  const int row0 = blockIdx.x * ROWS_PER_BLOCK;
  const float* gt_b = gt + (size_t)b * NQ * 4;

#if USE_TDM
  if (threadIdx.x < 32) {  // one TDM descriptor issued by wave 0
    const uint64_t ga = (uint64_t)(uintptr_t)gt_b;
    const unsigned lds_off = (unsigned)(uintptr_t)(void*)gtile;  // flat addr low 32b == LDS offset
    // D# group0: count=1, lds_addr, global_addr[56:0], type=2 ("image")
    u32x4 g0 = {1u, lds_off, (unsigned)ga,
                ((unsigned)((ga >> 32) & 0x01FFFFFFu)) | (2u << 30)};
    // D# group1: data_size=2 (4B), 1-D tile: tensor_dim0=tile_dim0=3600,
    // tensor_dim1=tile_dim1=1, dim0 stride = 3600.
    i32x8 g1 = {(int)0x00020000u,
                (int)(((unsigned)(NQ * 4) & 0xFFFFu) << 16),  // tensor_dim0 lo16 @ [63:48]
                (int)(1u << 16),                              // tensor_dim1 = 1
                (int)((unsigned)(NQ * 4) << 16),              // tile_dim0 = 3600
                1,                                            // tile_dim1 = 1
                NQ * 4, 0, 0};                                // tensor_dim0_stride
    i32x4 gz4 = {0, 0, 0, 0};
#if __clang_major__ >= 23
    i32x8 gz8 = {0, 0, 0, 0, 0, 0, 0, 0};
    __builtin_amdgcn_tensor_load_to_lds(g0, g1, gz4, gz4, gz8, 0);
#else
    __builtin_amdgcn_tensor_load_to_lds(g0, g1, gz4, gz4, 0);
#endif
#if __has_builtin(__builtin_amdgcn_s_wait_tensorcnt)
    __builtin_amdgcn_s_wait_tensorcnt(0);
#else
    asm volatile("s_wait_tensorcnt 0x0" ::: "memory");
#endif
    asm volatile("" ::: "memory");  // compiler barrier: LDS written by TDM
  }
#else
  for (int t = threadIdx.x; t < NQ * 4; t += blockDim.x) gtile[t] = gt_b[t];
#endif
  for (int t = threadIdx.x; t < NQ; t += blockDim.x) vtile[t] = valid[b * NQ + t];
  __syncthreads();

  for (int r = 0; r < ROWS_PER_BLOCK; ++r) {
    const int i = row0 + r;
    const float4 p = reinterpret_cast<const float4*>(pred)[(size_t)b * NQ + i];
    const float area_p = (p.z - p.x) * (p.w - p.y);
    const bool vi = vtile[i] != 0;
    const size_t rowbase = ((size_t)b * NQ + i) * NQ;
    for (int j = threadIdx.x; j < NQ; j += 256) {
      const float4 g = reinterpret_cast<const float4*>(gtile)[j];  // ds_load_b128
      const float area_g = (g.z - g.x) * (g.w - g.y);
      const float ltx = fmaxf(p.x, g.x), lty = fmaxf(p.y, g.y);
      const float rbx = fminf(p.z, g.z), rby = fminf(p.w, g.w);
      const float iw = fmaxf(rbx - ltx, 0.0f), ih = fmaxf(rby - lty, 0.0f);
      const float inter = iw * ih;
      const float uni = area_p + area_g - inter;
      const float iou = inter / (uni + GEPS);
      const float cx0 = fminf(p.x, g.x), cy0 = fminf(p.y, g.y);
      const float cx1 = fmaxf(p.z, g.z), cy1 = fmaxf(p.w, g.w);
      const float ew = fmaxf(cx1 - cx0, 0.0f), eh = fmaxf(cy1 - cy0, 0.0f);
      const float enc = ew * eh;
      const float giou = iou - (enc - uni) / (enc + GEPS);
      giou_ws[rowbase + j] = giou;  // default RT hint -> stays in 192MB L2
      const float mi = (vi && (vtile[j] != 0)) ? iou : -1.0f;
      __builtin_nontemporal_store(mi, iou_out + rowbase + j);  // stream to HBM
    }
  }
}

// ---------------------------------------------------------------------------
// Phase 2: exact Jonker-Volgenant shortest augmenting paths (minimize giou).
// One 1024-thread block (32 wave32 waves) per batch. Column-parallel relax +
// wave32 shuffle argmin; every 3.6KB cost-row scan hits the L2-resident
// cost tensor. Mirrors scipy/_lsa_min dual updates exactly.
// ---------------------------------------------------------------------------
__device__ __forceinline__ unsigned fkey(float f) {
  unsigned u = __float_as_uint(f);
  return (u >> 31) ? ~u : (u | 0x80000000u);
}
__device__ __forceinline__ float funkey(unsigned k) {
  return __uint_as_float((k >> 31) ? (k & 0x7FFFFFFFu) : ~k);
}
__device__ __forceinline__ unsigned long long wave_min_u64(unsigned long long k) {
  for (int off = 16; off > 0; off >>= 1) {
    unsigned long long o = __shfl_xor(k, off, 32);
    if (o < k) k = o;
  }
  return k;
}

__global__ __launch_bounds__(1024) void jv_solver_kernel(
    const float* __restrict__ giou, int* __restrict__ cols_out) {
  const int N = NQ;
  const int tid = threadIdx.x;
  const int lane = tid & 31, wid = tid >> 5;
  const float* cost = giou + (size_t)blockIdx.x * N * N;

  __shared__ float sp[NQ], u_s[NQ], v_s[NQ];
  __shared__ int path[NQ], col4row[NQ], row4col[NQ];
  __shared__ unsigned char SC[NQ], SR[NQ];
  __shared__ unsigned long long red[32];
  __shared__ int s_i, s_sink;
  __shared__ float s_minv;

  if (tid < N) { u_s[tid] = 0.0f; v_s[tid] = 0.0f; col4row[tid] = -1; row4col[tid] = -1; }
  __syncthreads();

  for (int cur = 0; cur < N; ++cur) {
    if (tid < N) { sp[tid] = __builtin_inff(); path[tid] = -1; SC[tid] = 0; SR[tid] = 0; }
    if (tid == 0) { s_i = cur; s_minv = 0.0f; s_sink = -1; }
    __syncthreads();

    while (true) {
      const int i = s_i;
      const float minv = s_minv;
      if (tid == 0) SR[i] = 1;
      if (tid < N) __builtin_prefetch(cost + (size_t)i * N + tid, 0, 0);  // global_prefetch_b8

      unsigned long long key = ~0ULL;
      if (tid < N && !SC[tid]) {
        float c = cost[(size_t)i * N + tid];
        c = (c == c) ? fminf(fmaxf(c, -100000.0f), 100000.0f) : 100000.0f;  // nan_to_num
        const float r = minv + c - u_s[i] - v_s[tid];
        if (r < sp[tid]) { sp[tid] = r; path[tid] = i; }
        key = ((unsigned long long)fkey(sp[tid]) << 32) | (unsigned)tid;  // ties -> lowest j
      }
      key = wave_min_u64(key);                // wave32 shuffle reduction
      if (lane == 0) red[wid] = key;
      __syncthreads();
      if (wid == 0) {
        unsigned long long k = red[lane];
        k = wave_min_u64(k);
        if (lane == 0) red[0] = k;
      }
      __syncthreads();
      const unsigned long long best = red[0];
      if (tid == 0) {
        const int j = (int)(best & 0xFFFFFFFFu);
        s_minv = funkey((unsigned)(best >> 32));
        SC[j] = 1;
        if (row4col[j] < 0) s_sink = j; else s_i = row4col[j];
      }
      __syncthreads();
      if (s_sink >= 0) break;
    }

    const float minv = s_minv;
    if (tid == 0) u_s[cur] += minv;
    if (tid < N && tid != cur && SR[tid]) u_s[tid] += minv - sp[col4row[tid]];
    if (tid < N && SC[tid]) v_s[tid] -= minv - sp[tid];
    __syncthreads();

    if (tid == 0) {  // augment along the stored path
      int j = s_sink;
      while (true) {
        const int ii = path[j];
        row4col[j] = ii;
        const int t = col4row[ii];
        col4row[ii] = j;
        j = t;
        if (ii == cur) break;
      }
    }
    __syncthreads();
  }

  if (tid < N) cols_out[blockIdx.x * N + tid] = col4row[tid];
}

// ---------------------------------------------------------------------------
// Phase 3: avg_loss = sum_b,i( -giou[b,i,cols[b,i]] ) / max(sum(valid),1)
// ---------------------------------------------------------------------------
__global__ __launch_bounds__(256) void finalize_kernel(
    const float* __restrict__ giou, const int* __restrict__ cols,
    const unsigned char* __restrict__ valid, float* __restrict__ out) {
  __shared__ float ssum[256];
  __shared__ int scnt[256];
  float s = 0.0f;
  int c = 0;
  for (int t = threadIdx.x; t < BATCH * NQ; t += 256) {
    const int j = cols[t];
    s -= giou[(size_t)t * NQ + j];
    c += valid[t] ? 1 : 0;
  }
  ssum[threadIdx.x] = s;
  scnt[threadIdx.x] = c;
  __syncthreads();
  for (int off = 128; off > 0; off >>= 1) {
    if (threadIdx.x < off) {
      ssum[threadIdx.x] += ssum[threadIdx.x + off];
      scnt[threadIdx.x] += scnt[threadIdx.x + off];
    }
    __syncthreads();
  }
  if (threadIdx.x == 0) out[0] = ssum[0] / fmaxf((float)scnt[0], 1.0f);
}

// ---------------------------------------------------------------------------
extern "C" void kernel_launch(void* const* d_in, const int* in_sizes, int n_in,
                              void* d_out, int out_size, void* d_ws, size_t ws_size,
                              hipStream_t stream) {
  (void)in_sizes; (void)n_in; (void)out_size; (void)ws_size;
  const float* pred = (const float*)d_in[0];
  const float* gt = (const float*)d_in[1];
  const unsigned char* valid = (const unsigned char*)d_in[2];  // bool mask
  float* out = (float*)d_out;                                  // [avg_loss | matched_ious]
  float* giou_ws = (float*)d_ws;                               // B*Q*Q floats (~104MB, L2-resident)
  int* col_ws = (int*)((char*)d_ws + (size_t)BATCH * NQ * NQ * sizeof(float));

  giou_pairwise_kernel<<<dim3(NQ / ROWS_PER_BLOCK, BATCH), 256, 0, stream>>>(
      pred, gt, valid, giou_ws, out + 1);
  jv_solver_kernel<<<BATCH, 1024, 0, stream>>>(giou_ws, col_ws);
  finalize_kernel<<<1, 256, 0, stream>>>(giou_ws, col_ws, valid, out);
}